// holtwinter_forecasting_46282567581933
// MI455X (gfx1250) — compile-verified
//
#include <hip/hip_runtime.h>

// Holt-Winters reformulated as: final (L, b, S19) are linear functionals of
// [x[1..511], init_L, init_b, init_S[0..23]] since the transition matrices
// depend only on (alpha,beta,gamma,t%24). Precompute coefficient vectors with
// a backward covector recurrence, then do a memory-bound skinny f32 GEMM with
// V_WMMA_F32_16X16X4_F32, plus a tiny per-window epilogue and reduction.

#define SEASONP 24
#define HORIZON 12
#define F_WINDOW 512
#define N_WINDOWS 32768

typedef __attribute__((ext_vector_type(2))) float v2f;
typedef __attribute__((ext_vector_type(8))) float v8f;

// Workspace layout (floats):
//   [0,512)     wL      coefficients on x for L_final
//   [512,1024)  wb      coefficients on x for b_final
//   [1024,1536) wS      coefficients on x for S19_final
//   [1536,1562) aL[26]  coefficients on [init_L, init_b, init_S[0..23]]
//   [1562,1588) ab[26]
//   [1588,1614) aS[26]

__global__ void hw_zero_kernel(float* out) {
    if (threadIdx.x == 0) out[0] = 0.0f;
}

// One wave; threads 0..2 each propagate one output component's covector
// backward through the 511 affine steps. r^{(t)} = r^{(t+1)} * A_t,
// w[t] = r^{(t+1)} . u_t.  State order: [L, b, S0..S23].
__global__ void hw_precompute_kernel(const float* __restrict__ alpha_p,
                                     const float* __restrict__ beta_p,
                                     const float* __restrict__ gamma_p,
                                     float* __restrict__ ws) {
    int c = threadIdx.x;
    if (c >= 3) return;
    const float a  = alpha_p[0];
    const float be = beta_p[0];
    const float g  = gamma_p[0];

    float r[26];
    for (int i = 0; i < 26; ++i) r[i] = 0.0f;
    if (c == 0)      r[0] = 1.0f;        // L_final
    else if (c == 1) r[1] = 1.0f;        // b_final
    else             r[2 + 19] = 1.0f;   // S19_final: (F-1+HORIZON)%P = 19

    float* wv = ws + c * F_WINDOW;
    wv[0] = 0.0f;                        // x[:,0] is unused by the scan

    const float ab_  = a * be;           // alpha*beta
    const float g1a  = g * (1.0f - a);   // gamma*(1-alpha)
    for (int t = F_WINDOW - 1; t >= 1; --t) {
        const int si = t % SEASONP;
        const float rL = r[0], rb = r[1], rS = r[2 + si];
        // w[t] = r . u_t,  u = [a, a*beta, g*(1-a)] at rows (L, b, S_si)
        wv[t] = rL * a + rb * ab_ + rS * g1a;
        // r <- r * A_t  (A_t is identity except columns L, b, S_si)
        const float nL = rL * (1.0f - a) + rb * (-ab_) + rS * (-g1a);
        const float nb = rL * (1.0f - a) + rb * (1.0f - ab_) + rS * (-g1a);
        const float nS = rL * (-a)       + rb * (-ab_) + rS * (1.0f - g1a);
        r[0] = nL; r[1] = nb; r[2 + si] = nS;
    }
    float* av = ws + 3 * F_WINDOW + c * 26;
    for (int i = 0; i < 26; ++i) av[i] = r[i];
}

// 256 threads = 8 waves; each wave computes a 16-window tile of the skinny
// GEMM  D(16x16) += A(16x4, windows x k) * B(4x16, k x coeff-cols) over
// 128 k-chunks, f32 WMMA. Only B columns 0..2 are live (wL, wb, wS).
__global__ __launch_bounds__(256) void hw_wmma_kernel(
    const float* __restrict__ x,      const float* __restrict__ y,
    const float* __restrict__ init_L, const float* __restrict__ init_b,
    const float* __restrict__ init_S, const float* __restrict__ ws,
    float* __restrict__ out) {
    __shared__ float tile[8][16][16];
    __shared__ float red[8];

    const int lane = threadIdx.x & 31;
    const int wave = threadIdx.x >> 5;
    const int col  = lane & 15;   // A row (window-in-tile) / B column
    const int half = lane >> 4;   // K sub-offset selector (0 -> k,k+1 ; 1 -> k+2,k+3)
    const int tileId = blockIdx.x * 8 + wave;
    const int win0 = tileId * 16;

    const float* xrow = x + (size_t)(win0 + col) * F_WINDOW + half * 2;
    // Branchless B source: lanes with col>=3 load a broadcast dummy and mask
    // to zero, keeping EXEC all-ones across the WMMA (ISA requirement).
    const int   ccol = (col < 3) ? col : 2;
    const float bmask = (col < 3) ? 1.0f : 0.0f;
    const float* wv = ws + ccol * F_WINDOW + half * 2;

    v8f acc = {0.f, 0.f, 0.f, 0.f, 0.f, 0.f, 0.f, 0.f};
    #pragma unroll 4
    for (int k = 0; k < F_WINDOW; k += 4) {
        v2f A = *(const v2f*)(xrow + k);
        v2f Bv = *(const v2f*)(wv + k);
        Bv.x *= bmask;
        Bv.y *= bmask;
        acc = __builtin_amdgcn_wmma_f32_16x16x4_f32(
            /*neg_a=*/false, A, /*neg_b=*/false, Bv,
            /*c_mod=*/(short)0, acc, /*reuse_a=*/false, /*reuse_b=*/false);
    }

    // D layout: lane = N + 16*(M>=8), vgpr = M%8  -> tile[wave][M][N]
    #pragma unroll
    for (int r = 0; r < 8; ++r)
        tile[wave][r + half * 8][col] = acc[r];
    __syncthreads();

    float partial = 0.0f;
    if (lane < 16) {
        const int w = win0 + lane;
        float Lf = tile[wave][lane][0];
        float bf = tile[wave][lane][1];
        float Sf = tile[wave][lane][2];
        const float* aL = ws + 3 * F_WINDOW;
        const float* ab = aL + 26;
        const float* aS = aL + 52;
        const float zL = init_L[w];
        const float zb = init_b[w];
        Lf += aL[0] * zL + aL[1] * zb;
        bf += ab[0] * zL + ab[1] * zb;
        Sf += aS[0] * zL + aS[1] * zb;
        const float* Sr = init_S + (size_t)w * SEASONP;
        #pragma unroll
        for (int j = 0; j < SEASONP; ++j) {
            const float s = Sr[j];
            Lf += aL[2 + j] * s;
            bf += ab[2 + j] * s;
            Sf += aS[2 + j] * s;
        }
        const float f = (Lf + (float)HORIZON * bf) * Sf;
        const float d = f - y[w];
        partial = d * d;
    }
    // wave reduction (wave32), then block reduction, one atomic per block
    #pragma unroll
    for (int off = 16; off > 0; off >>= 1)
        partial += __shfl_down(partial, off, 32);
    if (lane == 0) red[wave] = partial;
    __syncthreads();
    if (threadIdx.x == 0) {
        float s = 0.0f;
        #pragma unroll
        for (int i = 0; i < 8; ++i) s += red[i];
        atomicAdd(out, s);
    }
}

extern "C" void kernel_launch(void* const* d_in, const int* in_sizes, int n_in,
                              void* d_out, int out_size, void* d_ws, size_t ws_size,
                              hipStream_t stream) {
    const float* x      = (const float*)d_in[0];
    const float* y      = (const float*)d_in[1];
    const float* init_L = (const float*)d_in[2];
    const float* init_b = (const float*)d_in[3];
    const float* init_S = (const float*)d_in[4];
    const float* alpha  = (const float*)d_in[5];
    const float* beta   = (const float*)d_in[6];
    const float* gamma  = (const float*)d_in[7];
    float* ws  = (float*)d_ws;
    float* out = (float*)d_out;

    hw_zero_kernel<<<1, 32, 0, stream>>>(out);
    hw_precompute_kernel<<<1, 32, 0, stream>>>(alpha, beta, gamma, ws);
    // 32768 windows / (16 per wave * 8 waves per block) = 256 blocks
    hw_wmma_kernel<<<256, 256, 0, stream>>>(x, y, init_L, init_b, init_S, ws, out);
}